// ModelVoxel_46016279609477
// MI455X (gfx1250) — compile-verified
//
#include <hip/hip_runtime.h>

// Voxel nearest-neighbor gather:
//   out[p] = f[clip((x[p]+1)/h)] with h = 2/255, f is 256^3 f32 (64 MB).
//
// MI455X strategy (memory-bound, ~320 MB total traffic -> ~14us @ 23.3 TB/s):
//  - f (64 MB) fits in the 192 MB L2: keep it resident by marking the x-stream
//    loads and out-stream stores non-temporal (TH=NT) so they don't evict it.
//  - 4 points per thread: 3x aligned b128 NT loads of x, 4 scalar gathers from
//    f (regular temporal policy -> L2-cached), 1 b128 NT store of out.
//  - global_prefetch_b8 on the x stream ~48KB ahead (gfx1250 prefetch path).

typedef float v4f __attribute__((ext_vector_type(4)));

__device__ __forceinline__ int flat_idx(float px, float py, float pz) {
    // (x + 1) / h  ==  x * 127.5 + 127.5   (h = 2/255)
    constexpr float INV_H = 127.5f;
    float a = fminf(fmaxf(__builtin_fmaf(px, INV_H, INV_H), 0.0f), 255.0f);
    float b = fminf(fmaxf(__builtin_fmaf(py, INV_H, INV_H), 0.0f), 255.0f);
    float c = fminf(fmaxf(__builtin_fmaf(pz, INV_H, INV_H), 0.0f), 255.0f);
    // v_cvt_i32_f32 truncates toward zero; inputs are >= 0, matches astype(int32)
    return ((int)a << 16) | ((int)b << 8) | (int)c;   // dims are 256^3
}

__global__ __launch_bounds__(256) void voxel_gather_kernel(
    const float* __restrict__ x,
    const float* __restrict__ f,
    float* __restrict__ out,
    int npts) {
    const int t  = blockIdx.x * blockDim.x + threadIdx.x;
    const long p0 = (long)t * 4;                 // first point of this thread
    if (p0 >= npts) return;

    const v4f* xv = (const v4f*)(x + p0 * 3);    // 48*t bytes, 16B aligned

    // Prefetch the x stream well ahead (speculative, streaming side).
    __builtin_prefetch((const char*)xv + 49152, 0, 0);

    // Non-temporal: don't let the 192 MB x-stream evict f from L2.
    v4f a = __builtin_nontemporal_load(xv + 0);  // p0:(x,y,z) p1:(x)
    v4f b = __builtin_nontemporal_load(xv + 1);  // p1:(y,z)  p2:(x,y)
    v4f c = __builtin_nontemporal_load(xv + 2);  // p2:(z)    p3:(x,y,z)

    const int i0 = flat_idx(a.x, a.y, a.z);
    const int i1 = flat_idx(a.w, b.x, b.y);
    const int i2 = flat_idx(b.z, b.w, c.x);
    const int i3 = flat_idx(c.y, c.z, c.w);

    // Gathers: regular (temporal) loads so f stays L2-resident.
    v4f r;
    r.x = f[i0];
    r.y = f[i1];
    r.z = f[i2];
    r.w = f[i3];

    // Non-temporal store of the 64 MB output stream.
    __builtin_nontemporal_store(r, (v4f*)(out + p0));
}

extern "C" void kernel_launch(void* const* d_in, const int* in_sizes, int n_in,
                              void* d_out, int out_size, void* d_ws, size_t ws_size,
                              hipStream_t stream) {
    const float* x  = (const float*)d_in[0];   // (N_PTS, 3) f32
    const float* f  = (const float*)d_in[1];   // (256,256,256) f32
    float* out      = (float*)d_out;           // (N_PTS,) f32

    const int npts     = in_sizes[0] / 3;      // 16777216
    const int nthreads = (npts + 3) / 4;       // 4 points per thread
    const int block    = 256;                  // 8 waves (wave32) per block
    const int grid     = (nthreads + block - 1) / block;

    voxel_gather_kernel<<<grid, block, 0, stream>>>(x, f, out, npts);
}